// HungarianMatcherAVA_33681133535442
// MI455X (gfx1250) — compile-verified
//
#include <hip/hip_runtime.h>
#include <math.h>

// Problem constants (from setup_inputs): bs=128, nq=300, T=16.
#define BS 128
#define NQ 300
#define NT 16

#define W_CLASS 1.0f
#define W_BBOX  5.0f
#define W_GIOU  2.0f

// ---------------------------------------------------------------------------
// CDNA5 async global->LDS path (ASYNCcnt), guarded for toolchain portability.
// Builtin signature (from probe diagnostics): (as1 int*, as3 int*, Imm, Imm).
// ---------------------------------------------------------------------------
#if __has_builtin(__builtin_amdgcn_global_load_async_to_lds_b32)
#define HAVE_ASYNC_LDS 1
typedef __attribute__((address_space(1))) int as1_int;
typedef __attribute__((address_space(3))) int as3_int;
#define ASYNC_LDS_B32(gptr, lptr)                                              \
  __builtin_amdgcn_global_load_async_to_lds_b32((as1_int*)(void*)(gptr),       \
                                                (as3_int*)(lptr), 0, 0)
#else
#define HAVE_ASYNC_LDS 0
#endif

__device__ __forceinline__ void wait_async_zero() {
#if HAVE_ASYNC_LDS
#if __has_builtin(__builtin_amdgcn_s_wait_asynccnt)
  __builtin_amdgcn_s_wait_asynccnt(0);
#else
  asm volatile("s_wait_asynccnt 0" ::: "memory");
#endif
#endif
}

// ---------------------------------------------------------------------------
// Kernel 1: cost matrix  C[b][q][t] = 5*L1 + 2*(-GIoU) + 1*(-softmax_p1)
// One batch per block. Target boxes (16x5 = 80 dwords) staged into LDS via
// async-to-LDS loads; derived xyxy/area precomputed once.
// ---------------------------------------------------------------------------
__global__ __launch_bounds__(256)
void ava_cost_kernel(const float* __restrict__ pred_boxes,
                     const float* __restrict__ logits_b,
                     const float* __restrict__ tgt_boxes,
                     float* __restrict__ C) {
  const int b   = blockIdx.x;
  const int tid = threadIdx.x;

  __shared__ float traw[NT * 5];    // raw tgt rows: [cls, cx, cy, w, h]
  __shared__ float txyxy[NT * 4];
  __shared__ float tarea[NT];

  const float* tg = tgt_boxes + (size_t)b * NT * 5;
#if HAVE_ASYNC_LDS
  if (tid < NT * 5) ASYNC_LDS_B32(tg + tid, &traw[tid]);
  wait_async_zero();
#else
  if (tid < NT * 5) traw[tid] = tg[tid];
#endif
  __syncthreads();

  if (tid < NT) {
    float cx = traw[tid * 5 + 1], cy = traw[tid * 5 + 2];
    float w  = traw[tid * 5 + 3], h  = traw[tid * 5 + 4];
    txyxy[tid * 4 + 0] = cx - 0.5f * w;
    txyxy[tid * 4 + 1] = cy - 0.5f * h;
    txyxy[tid * 4 + 2] = cx + 0.5f * w;
    txyxy[tid * 4 + 3] = cy + 0.5f * h;
    tarea[tid] = w * h;
  }
  __syncthreads();

  for (int q = tid; q < NQ; q += blockDim.x) {
    const float4 p  = *(const float4*)(pred_boxes + ((size_t)b * NQ + q) * 4);
    const float2 lg = *(const float2*)(logits_b   + ((size_t)b * NQ + q) * 2);
    // softmax over 2 classes, keep class-1 prob
    float p1    = 1.0f / (1.0f + expf(lg.x - lg.y));
    float ccls  = -p1;
    float ax1 = p.x - 0.5f * p.z, ay1 = p.y - 0.5f * p.w;
    float ax2 = p.x + 0.5f * p.z, ay2 = p.y + 0.5f * p.w;
    float areaA = p.z * p.w;

    float* out = C + ((size_t)b * NQ + q) * NT;
#pragma unroll
    for (int t = 0; t < NT; ++t) {
      float l1 = fabsf(p.x - traw[t * 5 + 1]) + fabsf(p.y - traw[t * 5 + 2]) +
                 fabsf(p.z - traw[t * 5 + 3]) + fabsf(p.w - traw[t * 5 + 4]);
      float bx1 = txyxy[t * 4 + 0], by1 = txyxy[t * 4 + 1];
      float bx2 = txyxy[t * 4 + 2], by2 = txyxy[t * 4 + 3];
      float iw = fmaxf(fminf(ax2, bx2) - fmaxf(ax1, bx1), 0.0f);
      float ih = fmaxf(fminf(ay2, by2) - fmaxf(ay1, by1), 0.0f);
      float inter = iw * ih;
      float uni   = areaA + tarea[t] - inter;
      float iou   = inter / uni;
      float ew = fmaxf(fmaxf(ax2, bx2) - fminf(ax1, bx1), 0.0f);
      float eh = fmaxf(fmaxf(ay2, by2) - fminf(ay1, by1), 0.0f);
      float areaE = ew * eh;
      float giou  = iou - (areaE - uni) / areaE;
      out[t] = W_BBOX * l1 + W_GIOU * (-giou) + W_CLASS * ccls;
    }
  }
}

// ---------------------------------------------------------------------------
// Kernel 2: Jonker-Volgenant shortest-augmenting-path LSA on the transposed
// 16x300 matrix. One batch per block, one wave32 per block. The 300-column
// Dijkstra scan + argmin are lane-parallel; O(16) bookkeeping on lane 0.
// Cost tile is transposed into LDS using per-lane-scatter async-to-LDS loads.
// ---------------------------------------------------------------------------
__global__ __launch_bounds__(32)
void ava_hungarian_kernel(const float* __restrict__ C,
                          float* __restrict__ rows_out,
                          float* __restrict__ cols_out) {
  const int b    = blockIdx.x;
  const int lane = threadIdx.x;

  __shared__ float costT[NT * NQ];   // transposed: costT[t][q]
  __shared__ float minv[NQ];
  __shared__ float vdual[NQ];
  __shared__ float udual[NT];
  __shared__ int   path[NQ];
  __shared__ int   row4col[NQ];
  __shared__ int   remaining[NQ];
  __shared__ int   col4row[NT];
  __shared__ int   SR[NT];
  __shared__ int   SRn;

  const float* Cb = C + (size_t)b * NQ * NT;

  // Prefetch the 19.2KB tile (75 x 256B lines) into cache (global_prefetch_b8).
  for (int line = lane; line < (NQ * NT * 4) / 256; line += 32)
    __builtin_prefetch(Cb + line * 64, 0, 3);

  // Stage + transpose: lane reads C[b][q][t] contiguously, scatters to LDS
  // at costT[t*NQ+q] (async-to-LDS takes a per-lane LDS destination VGPR).
  for (int n = lane; n < NQ * NT; n += 32) {
    int q = n >> 4, t = n & (NT - 1);
#if HAVE_ASYNC_LDS
    ASYNC_LDS_B32(Cb + n, &costT[t * NQ + q]);
#else
    costT[t * NQ + q] = Cb[n];
#endif
  }
  if (lane < NT) { udual[lane] = 0.0f; col4row[lane] = -1; }
  for (int j = lane; j < NQ; j += 32) { vdual[j] = 0.0f; row4col[j] = -1; }
  wait_async_zero();
  __syncthreads();

  const float INFF = __builtin_inff();

  for (int cur_row = 0; cur_row < NT; ++cur_row) {
    for (int j = lane; j < NQ; j += 32) {
      minv[j] = INFF; path[j] = -1; remaining[j] = 1;
    }
    if (lane == 0) { SRn = 1; SR[0] = cur_row; }
    __syncthreads();

    int   i        = cur_row;
    float shortest = 0.0f;
    int   sink     = -1;

    while (sink < 0) {
      float ui = udual[i];
      const float* crow = &costT[i * NQ];
      for (int j = lane; j < NQ; j += 32) {
        if (remaining[j]) {
          float r = shortest + crow[j] - ui - vdual[j];
          if (r < minv[j]) { minv[j] = r; path[j] = i; }
        }
      }
      __syncthreads();

      // argmin over remaining columns; tie-break lowest index (numpy argmin).
      float mv = INFF; int mi = 0x7fffffff;
      for (int j = lane; j < NQ; j += 32) {
        if (remaining[j]) {
          float m = minv[j];
          if (m < mv || (m == mv && j < mi)) { mv = m; mi = j; }
        }
      }
      for (int off = 16; off > 0; off >>= 1) {
        float ov = __shfl_xor(mv, off, 32);
        int   oi = __shfl_xor(mi, off, 32);
        if (ov < mv || (ov == mv && oi < mi)) { mv = ov; mi = oi; }
      }
      int k = mi;
      shortest = mv;
      if (lane == 0) remaining[k] = 0;
      __syncthreads();

      int rk = row4col[k];           // uniform LDS read -> uniform control
      if (rk < 0) {
        sink = k;
      } else {
        i = rk;
        if (lane == 0) { SR[SRn] = rk; SRn = SRn + 1; }
      }
      __syncthreads();
    }

    // Dual updates (lane 0: O(16); lanes: O(300) v-update), then augment.
    if (lane == 0) {
      udual[cur_row] += shortest;
      for (int s = 1; s < SRn; ++s) {
        int ii = SR[s];
        udual[ii] += shortest - minv[col4row[ii]];
      }
    }
    __syncthreads();
    for (int j = lane; j < NQ; j += 32) {
      if (!remaining[j]) vdual[j] -= shortest - minv[j];
    }
    if (lane == 0) {
      int j = sink;
      for (;;) {
        int ii = path[j];
        row4col[j] = ii;
        int nj = col4row[ii];
        col4row[ii] = j;
        j = nj;
        if (ii == cur_row) break;
      }
    }
    __syncthreads();
  }

  // order = argsort(col4row); rows = col4row[order]; cols = order.
  // 16 distinct values -> rank = count-less-than (exact numpy ascending sort).
  if (lane < NT) {
    int myc  = col4row[lane];
    int rank = 0;
#pragma unroll
    for (int s = 0; s < NT; ++s) rank += (col4row[s] < myc) ? 1 : 0;
    rows_out[b * NT + rank] = (float)myc;
    cols_out[b * NT + rank] = (float)lane;
  }
}

// ---------------------------------------------------------------------------
extern "C" void kernel_launch(void* const* d_in, const int* in_sizes, int n_in,
                              void* d_out, int out_size, void* d_ws, size_t ws_size,
                              hipStream_t stream) {
  (void)in_sizes; (void)n_in; (void)out_size; (void)d_ws; (void)ws_size;
  // setup_inputs order: pred_logits (unused), pred_boxes, pred_logits_b, tgt_boxes
  const float* pred_boxes = (const float*)d_in[1];
  const float* logits_b   = (const float*)d_in[2];
  const float* tgt_boxes  = (const float*)d_in[3];

  float* C        = (float*)d_out;                    // 128*300*16 floats
  float* rows_out = C + (size_t)BS * NQ * NT;         // 128*16 floats
  float* cols_out = rows_out + (size_t)BS * NT;       // 128*16 floats

  ava_cost_kernel<<<BS, 256, 0, stream>>>(pred_boxes, logits_b, tgt_boxes, C);
  ava_hungarian_kernel<<<BS, 32, 0, stream>>>(C, rows_out, cols_out);
}